// CGaANLayer_34883724378281
// MI455X (gfx1250) — compile-verified
//
#include <hip/hip_runtime.h>
#include <hip/hip_bf16.h>
#include <math.h>

// ---------------------------------------------------------------------------
// CGaAN layer for MI455X (gfx1250, wave32).
//   z = v@Wa ; gproj = v@Wgm           -> V_WMMA_F32_16X16X4_F32 (fp32 exact)
//   segment softmax / sum / max / mean -> L2-resident f32 atomics (192MB L2
//   holds the whole ~60MB working set, so edge phase is L2-bound, not HBM)
// Softmax normalization is deferred to the node epilogue (h_accum/esum),
// which fuses the exp+sum pass with the wide accumulation pass: only TWO
// edge sweeps total instead of three.
// ---------------------------------------------------------------------------

typedef __attribute__((ext_vector_type(2))) float v2f;
typedef __attribute__((ext_vector_type(8))) float v8f;

#define DIM 64            // IN_DIM == OUT_DIM == 64

// ----- float atomic max via ordered-int trick (global_atomic_max/min) -------
__device__ __forceinline__ void atomicMaxF32(float* addr, float val) {
  // Non-negative floats order like signed ints; negative floats order
  // reversed as unsigned ints. Mixed-sign races still converge to the max.
  if (!(val < 0.0f)) {
    atomicMax((int*)addr, __float_as_int(val));
  } else {
    atomicMin((unsigned int*)addr, __float_as_uint(val));
  }
}

// ----- 0: initialize accumulators (workspace is poisoned; re-init per call) -
__global__ void init_kernel(float* __restrict__ h, float* __restrict__ gmax,
                            float* __restrict__ vsum, float* __restrict__ emax,
                            float* __restrict__ esum, float* __restrict__ deg,
                            int n64, int n) {
  int i = blockIdx.x * blockDim.x + threadIdx.x;
  if (i < n64) {
    h[i] = 0.0f;
    gmax[i] = -INFINITY;
    vsum[i] = 0.0f;
  }
  if (i < n) {
    emax[i] = -INFINITY;
    esum[i] = 0.0f;
    deg[i] = 0.0f;
  }
}

// ----- 1: fused projection GEMM via WMMA ------------------------------------
// Block = 256 threads = 8 waves; block owns a 16-row tile of v.
// Waves 0..3 -> z = v@Wa N-tiles 0..3 ; waves 4..7 -> gproj = v@Wgm.
// K=64 consumed as 16 chained V_WMMA_F32_16X16X4_F32 (compiler pipelines
// the fragment loads across the chain).
__global__ __launch_bounds__(256) void proj_wmma_kernel(
    const float* __restrict__ v, const float* __restrict__ Wa,
    const float* __restrict__ Wgm, float* __restrict__ z,
    float* __restrict__ gproj, int N) {
  const int lane = threadIdx.x & 31;
  const int wave = threadIdx.x >> 5;            // 0..7
  const int m0 = blockIdx.x * 16;               // row tile
  const int n0 = (wave & 3) * 16;               // col tile
  const float* __restrict__ W = (wave < 4) ? Wa : Wgm;
  float* __restrict__ D = (wave < 4) ? z : gproj;

  const int half = lane >> 4;                   // 0: lanes 0-15, 1: 16-31
  const int lmod = lane & 15;
  int arow = m0 + lmod;
  if (arow >= N) arow = N - 1;                  // clamp (stores are guarded)
  const float* __restrict__ vrow = v + (size_t)arow * DIM;

  v8f acc = {};
#pragma unroll
  for (int k0 = 0; k0 < DIM; k0 += 4) {
    // A fragment (16x4, ISA 7.12.2): half 0 holds K=k0,k0+1; half 1 K=k0+2,3
    v2f a;
    a.x = vrow[k0 + 2 * half + 0];
    a.y = vrow[k0 + 2 * half + 1];
    // B fragment (4x16), mirrored layout: VGPR j, half h -> K=k0+j+2h, N=n0+lmod
    v2f b;
    b.x = W[(size_t)(k0 + 2 * half + 0) * DIM + n0 + lmod];
    b.y = W[(size_t)(k0 + 2 * half + 1) * DIM + n0 + lmod];
    acc = __builtin_amdgcn_wmma_f32_16x16x4_f32(
        /*neg_a=*/false, a, /*neg_b=*/false, b,
        /*c_mod=*/(short)0, acc, /*reuse_a=*/false, /*reuse_b=*/false);
  }
  // D layout (16x16 f32 C/D): VGPR r, lane -> row m0 + r + 8*half, col n0+lmod
#pragma unroll
  for (int r = 0; r < 8; ++r) {
    int row = m0 + r + 8 * half;
    if (row < N) D[(size_t)row * DIM + n0 + lmod] = acc[r];
  }
}

// ----- 2: per-node attention scalars el = z.att_l, er = z.att_r -------------
__global__ __launch_bounds__(256) void node_att_kernel(
    const float* __restrict__ z, const float* __restrict__ att_l,
    const float* __restrict__ att_r, float* __restrict__ el,
    float* __restrict__ er, int N) {
  int node = (blockIdx.x * blockDim.x + threadIdx.x) >> 5;
  int lane = threadIdx.x & 31;
  if (node >= N) return;
  float2 zv = ((const float2*)(z + (size_t)node * DIM))[lane];
  float2 al = ((const float2*)att_l)[lane];
  float2 ar = ((const float2*)att_r)[lane];
  float pl = zv.x * al.x + zv.y * al.y;
  float pr = zv.x * ar.x + zv.y * ar.y;
#pragma unroll
  for (int off = 16; off > 0; off >>= 1) {
    pl += __shfl_xor(pl, off, 32);
    pr += __shfl_xor(pr, off, 32);
  }
  if (lane == 0) {
    el[node] = pl;
    er[node] = pr;
  }
}

// ----- 3: edge pass 1: leaky-relu logits, segment max, degree ---------------
__global__ void edge_pass1(const int* __restrict__ src,
                           const int* __restrict__ dst,
                           const float* __restrict__ el,
                           const float* __restrict__ er,
                           float* __restrict__ ebuf, float* __restrict__ emax,
                           float* __restrict__ deg, int E) {
  int i = blockIdx.x * blockDim.x + threadIdx.x;
  if (i >= E) return;
  int s = src[i], d = dst[i];
  float e = el[s] + er[d];
  e = (e > 0.0f) ? e : 0.01f * e;          // leaky_relu, slope 0.01
  ebuf[i] = e;
  atomicMaxF32(emax + d, e);
  atomicAdd(deg + d, 1.0f);
}

// ----- 4: fused wide edge pass: exp + esum + h_accum + gmax + vsum ----------
// One wave32 per edge; each lane owns 2 of the 64 dims (coalesced float2).
// h accumulates UNNORMALIZED ex*z[src]; division by esum happens per-node in
// the epilogue, so esum does not need to be complete before this pass.
__global__ __launch_bounds__(256) void edge_wide(
    const int* __restrict__ src, const int* __restrict__ dst,
    const float* __restrict__ ebuf, const float* __restrict__ emax,
    const float* __restrict__ z, const float* __restrict__ gproj,
    const float* __restrict__ v, float* __restrict__ esum,
    float* __restrict__ h, float* __restrict__ gmax,
    float* __restrict__ vsum, int E) {
  int g = blockIdx.x * blockDim.x + threadIdx.x;
  int eidx = g >> 5;
  int lane = g & 31;
  if (eidx >= E) return;
  int s = src[eidx], d = dst[eidx];
  // emax[d] is finite for every d that appears as an edge destination.
  float ex = __expf(ebuf[eidx] - emax[d]);   // wave-uniform
  if (lane == 0) atomicAdd(esum + d, ex);
  size_t so = (size_t)s * DIM + lane * 2;
  size_t doff = (size_t)d * DIM + lane * 2;
  float2 zv = *(const float2*)(z + so);
  float2 gv = *(const float2*)(gproj + so);
  float2 vv = *(const float2*)(v + so);
  atomicAdd(h + doff + 0, ex * zv.x);
  atomicAdd(h + doff + 1, ex * zv.y);
  atomicMaxF32(gmax + doff + 0, gv.x);
  atomicMaxF32(gmax + doff + 1, gv.y);
  atomicAdd(vsum + doff + 0, vv.x);
  atomicAdd(vsum + doff + 1, vv.y);
}

// ----- 5: node epilogue: softmax-normalize h, gate, residual ----------------
__global__ __launch_bounds__(256) void node_final(
    const float* __restrict__ v, const float* __restrict__ z,
    const float* __restrict__ h, const float* __restrict__ gmax,
    const float* __restrict__ vsum, const float* __restrict__ deg,
    const float* __restrict__ esum, const float* __restrict__ gate_l,
    const float* __restrict__ gate_m, const float* __restrict__ gate_r,
    float* __restrict__ out, int N) {
  int node = (blockIdx.x * blockDim.x + threadIdx.x) >> 5;
  int lane = threadIdx.x & 31;
  if (node >= N) return;
  size_t off = (size_t)node * DIM + lane * 2;
  float dg = deg[node];
  float2 vv = *(const float2*)(v + off);
  float2 gm;
  if (dg > 0.0f) {
    gm = *(const float2*)(gmax + off);
  } else {
    gm.x = 0.0f;
    gm.y = 0.0f;
  }
  float2 vs = *(const float2*)(vsum + off);
  float inv = 1.0f / fmaxf(dg, 1.0f);
  float2 gl = ((const float2*)gate_l)[lane];
  float2 gmw = ((const float2*)gate_m)[lane];
  float2 gr = ((const float2*)gate_r)[lane];
  float p = vv.x * gl.x + vv.y * gl.y + gm.x * gmw.x + gm.y * gmw.y +
            (vs.x * inv) * gr.x + (vs.y * inv) * gr.y;
#pragma unroll
  for (int o = 16; o > 0; o >>= 1) p += __shfl_xor(p, o, 32);
  float gval = 1.0f / (1.0f + __expf(-p));
  float es = esum[node];
  float hscale = (es > 0.0f) ? (gval / es) : 0.0f;  // deferred softmax norm
  float2 zv = *(const float2*)(z + off);
  float2 hv = *(const float2*)(h + off);
  float2 res;
  res.x = zv.x + hscale * hv.x;
  res.y = zv.y + hscale * hv.y;
  *(float2*)(out + off) = res;
}

// ---------------------------------------------------------------------------
extern "C" void kernel_launch(void* const* d_in, const int* in_sizes, int n_in,
                              void* d_out, int out_size, void* d_ws,
                              size_t ws_size, hipStream_t stream) {
  const float* v      = (const float*)d_in[0];
  const int*   src    = (const int*)d_in[1];
  const int*   dst    = (const int*)d_in[2];
  const float* Wa     = (const float*)d_in[3];
  const float* att_l  = (const float*)d_in[4];
  const float* att_r  = (const float*)d_in[5];
  const float* gate_l = (const float*)d_in[6];
  const float* gate_m = (const float*)d_in[7];
  const float* gate_r = (const float*)d_in[8];
  const float* Wgm    = (const float*)d_in[9];
  float* out = (float*)d_out;

  const int N = in_sizes[0] / DIM;   // 50000
  const int E = in_sizes[1];         // 800000
  const size_t N64 = (size_t)N * DIM;

  // workspace carve-up (~68 MB of f32)
  float* ws    = (float*)d_ws;
  float* z     = ws;            // N*64
  float* gproj = z + N64;       // N*64
  float* h     = gproj + N64;   // N*64
  float* gmaxb = h + N64;       // N*64
  float* vsum  = gmaxb + N64;   // N*64
  float* el    = vsum + N64;    // N
  float* er    = el + N;        // N
  float* emax  = er + N;        // N
  float* esum  = emax + N;      // N
  float* deg   = esum + N;      // N
  float* ebuf  = deg + N;       // E

  const int T = 256;

  // 0: init accumulators
  {
    int work = (int)N64;  // covers both n64 and n ranges
    init_kernel<<<(work + T - 1) / T, T, 0, stream>>>(h, gmaxb, vsum, emax,
                                                      esum, deg, work, N);
  }
  // 1: WMMA projections
  {
    int rowTiles = (N + 15) / 16;
    proj_wmma_kernel<<<rowTiles, T, 0, stream>>>(v, Wa, Wgm, z, gproj, N);
  }
  // 2: el / er
  node_att_kernel<<<(N * 32 + T - 1) / T, T, 0, stream>>>(z, att_l, att_r, el,
                                                          er, N);
  // 3: edge logits + segment max + degree
  edge_pass1<<<(E + T - 1) / T, T, 0, stream>>>(src, dst, el, er, ebuf, emax,
                                                deg, E);
  // 4: fused wide edge pass (wave per edge)
  {
    long long threads = (long long)E * 32;
    edge_wide<<<(unsigned)((threads + T - 1) / T), T, 0, stream>>>(
        src, dst, ebuf, emax, z, gproj, v, esum, h, gmaxb, vsum, E);
  }
  // 5: gate + residual epilogue (includes deferred softmax normalization)
  node_final<<<(N * 32 + T - 1) / T, T, 0, stream>>>(
      v, z, h, gmaxb, vsum, deg, esum, gate_l, gate_m, gate_r, out, N);
}